// BestRqFramework_28475633172776
// MI455X (gfx1250) — compile-verified
//
#include <hip/hip_runtime.h>
#include <hip/hip_bf16.h>

typedef __attribute__((ext_vector_type(2))) float v2f;
typedef __attribute__((ext_vector_type(8))) float v8f;

#define D_DIM 256
#define Q_DIM 64
#define L_DIM 2048
#define B_DIM 8
#define NROWS (B_DIM * L_DIM)              // 16384
#define NTILES (NROWS / 16)                // 1024 (16-row WMMA tiles)
#define WAVES_PER_BLOCK 4
#define NBLOCKS (NTILES / WAVES_PER_BLOCK) // 256

// workspace layout (float offsets); total < 25 KB
#define WS_PARTIALS 0      // 1024 tiles * {sum, sumsq}
#define WS_MU       2048   // 8
#define WS_RSTD     2056   // 8
#define WS_CBNT     2080   // 4096 floats, transposed normalized codebook [j*64 + i]

// One wave computes a 16(M) x 64(N) fp32 tile of t = x * W^T using
// v_wmma_f32_16x16x4_f32, K-loop over D=256 in steps of 4.
// A (16x4 f32): lane = {m = lane&15, kh = lane>>4}; VGPR0/1 = x[m][k0+2kh+{0,1}]
// B (4x16 f32): lane supplies W[n = nt*16 + m][k0+2kh+{0,1}]  (B[k][n] = W[n][k])
static __device__ __forceinline__ void gemm_tile(const float* __restrict__ x,
                                                 const float* __restrict__ W,
                                                 int row0, int m, int kh,
                                                 v8f acc[4]) {
  const float* xrow = x + (size_t)(row0 + m) * D_DIM + 2 * kh;
  const float* wrow = W + (size_t)m * D_DIM + 2 * kh;
#pragma unroll 4
  for (int k0 = 0; k0 < D_DIM; k0 += 4) {
    v2f a  = *(const v2f*)(xrow + k0);
    v2f b0 = *(const v2f*)(wrow + k0);
    v2f b1 = *(const v2f*)(wrow + k0 + 16 * D_DIM);
    v2f b2 = *(const v2f*)(wrow + k0 + 32 * D_DIM);
    v2f b3 = *(const v2f*)(wrow + k0 + 48 * D_DIM);
    acc[0] = __builtin_amdgcn_wmma_f32_16x16x4_f32(false, a, false, b0, (short)0, acc[0], false, false);
    acc[1] = __builtin_amdgcn_wmma_f32_16x16x4_f32(false, a, false, b1, (short)0, acc[1], false, false);
    acc[2] = __builtin_amdgcn_wmma_f32_16x16x4_f32(false, a, false, b2, (short)0, acc[2], false, false);
    acc[3] = __builtin_amdgcn_wmma_f32_16x16x4_f32(false, a, false, b3, (short)0, acc[3], false, false);
  }
}

// K1: GEMM tile + deterministic per-tile sum/sumsq partials
__global__ __launch_bounds__(128) void k1_gemm_stats(const float* __restrict__ x,
                                                     const float* __restrict__ W,
                                                     float* __restrict__ ws) {
  const int lane = threadIdx.x & 31;
  const int wave = threadIdx.x >> 5;
  const int tile = blockIdx.x * WAVES_PER_BLOCK + wave;
  const int row0 = tile * 16;
  const int m = lane & 15, kh = lane >> 4;

  v8f acc[4] = {};
  gemm_tile(x, W, row0, m, kh, acc);

  float s = 0.f, sq = 0.f;
#pragma unroll
  for (int nt = 0; nt < 4; ++nt)
#pragma unroll
    for (int r = 0; r < 8; ++r) {
      float v = acc[nt][r];
      s += v;
      sq += v * v;
    }
#pragma unroll
  for (int off = 16; off > 0; off >>= 1) {
    s  += __shfl_xor(s, off, 32);
    sq += __shfl_xor(sq, off, 32);
  }
  if (lane == 0) {
    ws[WS_PARTIALS + tile * 2]     = s;
    ws[WS_PARTIALS + tile * 2 + 1] = sq;
  }
}

// K2: reduce partials -> per-batch mu/rstd; normalize codebook (transposed)
__global__ __launch_bounds__(256) void k2_finalize(const float* __restrict__ cb,
                                                   float* __restrict__ ws) {
  __shared__ float r1[256], r2[256];
  __shared__ float cbmu_s, cbrstd_s;
  const int t = threadIdx.x;

  for (int b = 0; b < B_DIM; ++b) {
    float s = 0.f, sq = 0.f;
    if (t < 128) {
      s  = ws[WS_PARTIALS + (b * 128 + t) * 2];
      sq = ws[WS_PARTIALS + (b * 128 + t) * 2 + 1];
    }
    r1[t] = s; r2[t] = sq;
    __syncthreads();
    for (int off = 64; off > 0; off >>= 1) {
      if (t < off) { r1[t] += r1[t + off]; r2[t] += r2[t + off]; }
      __syncthreads();
    }
    if (t == 0) {
      const float inv_n = 1.0f / (float)(L_DIM * Q_DIM);
      float mu  = r1[0] * inv_n;
      float var = r2[0] * inv_n - mu * mu;
      ws[WS_MU + b]   = mu;
      ws[WS_RSTD + b] = rsqrtf(var + 1e-5f);
    }
    __syncthreads();
  }

  // codebook stats over 64*64 = 4096 elements
  float s = 0.f, sq = 0.f;
#pragma unroll
  for (int k = 0; k < 16; ++k) {
    float v = cb[t + 256 * k];
    s += v; sq += v * v;
  }
  r1[t] = s; r2[t] = sq;
  __syncthreads();
  for (int off = 128; off > 0; off >>= 1) {
    if (t < off) { r1[t] += r1[t + off]; r2[t] += r2[t + off]; }
    __syncthreads();
  }
  if (t == 0) {
    const float inv_n = 1.0f / 4096.0f;
    float mu  = r1[0] * inv_n;
    float var = r2[0] * inv_n - mu * mu;
    cbmu_s   = mu;
    cbrstd_s = rsqrtf(var + 1e-5f);
  }
  __syncthreads();
  const float cbmu = cbmu_s, cbrstd = cbrstd_s;
#pragma unroll
  for (int k = 0; k < 16; ++k) {
    int idx = t + 256 * k;
    int i = idx >> 6, j = idx & 63;
    ws[WS_CBNT + j * 64 + i] = (cb[idx] - cbmu) * cbrstd;  // transposed store
  }
}

// K3: recompute tile, normalize, first-min argmin over j, write int32 labels
__global__ __launch_bounds__(128) void k3_labels(const float* __restrict__ x,
                                                 const float* __restrict__ W,
                                                 const float* __restrict__ ws,
                                                 int* __restrict__ out) {
  __shared__ float cbnT[4096];  // [j*64 + i]
  {
    const float4* src = (const float4*)(ws + WS_CBNT);
    float4* dst = (float4*)cbnT;
#pragma unroll
    for (int k = 0; k < 8; ++k) dst[threadIdx.x + 128 * k] = src[threadIdx.x + 128 * k];
  }
  __syncthreads();

  const int lane = threadIdx.x & 31;
  const int wave = threadIdx.x >> 5;
  const int tile = blockIdx.x * WAVES_PER_BLOCK + wave;
  const int row0 = tile * 16;           // tile never straddles a batch (2048 % 16 == 0)
  const int m = lane & 15, kh = lane >> 4;
  const int b = row0 >> 11;             // row0 / L_DIM
  const float mu   = ws[WS_MU + b];
  const float rstd = ws[WS_RSTD + b];

  v8f acc[4] = {};
  gemm_tile(x, W, row0, m, kh, acc);

#pragma unroll
  for (int nt = 0; nt < 4; ++nt) {
    const int i = nt * 16 + m;
#pragma unroll
    for (int r = 0; r < 8; ++r) {
      const float tn = (acc[nt][r] - mu) * rstd;
      float best = 3.402823466e38f;
      int bj = 0;
      for (int j = 0; j < 64; ++j) {
        float dv = tn - cbnT[j * 64 + i];   // lanes hit distinct banks (stride 1 in i)
        if (dv < best) { best = dv; bj = j; }  // strict < keeps first minimum (jnp.argmin)
      }
      out[(size_t)(row0 + r + 8 * kh) * Q_DIM + i] = bj;
    }
  }
}

extern "C" void kernel_launch(void* const* d_in, const int* in_sizes, int n_in,
                              void* d_out, int out_size, void* d_ws, size_t ws_size,
                              hipStream_t stream) {
  const float* x  = (const float*)d_in[0];  // (8, 2048, 256)
  const float* W  = (const float*)d_in[1];  // (64, 256)
  const float* cb = (const float*)d_in[2];  // (64, 64)
  float* ws = (float*)d_ws;
  int* out  = (int*)d_out;                  // labels (8, 2048, 64) int32

  k1_gemm_stats<<<NBLOCKS, 128, 0, stream>>>(x, W, ws);
  k2_finalize<<<1, 256, 0, stream>>>(cb, ws);
  k3_labels<<<NBLOCKS, 128, 0, stream>>>(x, W, ws, out);
}